// VectorQuantizeEMA_10127532884559
// MI455X (gfx1250) — compile-verified
//
#include <hip/hip_runtime.h>

typedef float v2f __attribute__((ext_vector_type(2)));
typedef float v8f __attribute__((ext_vector_type(8)));

#define NUM_EMB   1024
#define EMB_DIM   64
#define JD        1024            // 32*32 spatial positions per batch
#define T_TOTAL   32768           // 32 batches * 1024
#define Q_ELEMS   2097152
#define ENC_ELEMS 33554432
#define LOSS_OFF  (Q_ELEMS)
#define PERP_OFF  (Q_ELEMS + 1)
#define ENC_OFF   (Q_ELEMS + 2)
#define DIST_OFF  (ENC_OFF + ENC_ELEMS)

// ---------------------------------------------------------------------------
// 1) Row/code squared norms + zero the accumulators (re-run every launch so
//    graph replay stays deterministic).
// ---------------------------------------------------------------------------
__global__ void __launch_bounds__(256)
vq_prep(const float* __restrict__ latent, const float* __restrict__ W,
        float* __restrict__ xsq, float* __restrict__ wsq,
        int* __restrict__ counts, float* __restrict__ lossacc)
{
    int t = blockIdx.x * 256 + threadIdx.x;
    if (t < T_TOTAL) {
        int b = t >> 10, jd = t & 1023;
        const float* base = latent + (size_t)b * (EMB_DIM * JD) + jd;
        float s = 0.f;
        #pragma unroll
        for (int n = 0; n < EMB_DIM; ++n) { float v = base[(size_t)n * JD]; s += v * v; }
        xsq[t] = s;
    }
    if (t < NUM_EMB) {
        const float* w = W + (size_t)t * EMB_DIM;
        float s = 0.f;
        #pragma unroll
        for (int n = 0; n < EMB_DIM; ++n) { float v = w[n]; s += v * v; }
        wsq[t] = s;
        counts[t] = 0;
    }
    if (t == 0) *lossacc = 0.f;
}

// ---------------------------------------------------------------------------
// 2) Fused distance GEMM (V_WMMA_F32_16X16X4_F32) + row-wise argmin.
//    Block = 8 waves; block owns 16 rows x all 1024 codes; wave owns 8
//    N-tiles of 16. Distances are written (they are an output) but the
//    argmin consumes the register values directly -> no 128 MB re-read.
//    dist[t][k] = ||x_t||^2 + ||w_k||^2 - 2 * (X @ W^T)[t][k]
// ---------------------------------------------------------------------------
__global__ void __launch_bounds__(256)
vq_dist(const float* __restrict__ latent, const float* __restrict__ W,
        const float* __restrict__ xsq, const float* __restrict__ wsq,
        float* __restrict__ dist, int* __restrict__ idx,
        int* __restrict__ counts)
{
    __shared__ float lmin[16][8];
    __shared__ int   lidx[16][8];

    const int lane  = threadIdx.x & 31;
    const int wave  = threadIdx.x >> 5;
    const int t0    = blockIdx.x * 16;         // 16 rows per block (same batch b)
    const int b     = t0 >> 10;
    const int jd0   = t0 & 1023;
    const int m     = lane & 15;               // A-matrix row for this lane
    const int khalf = lane >> 4;               // 0: K={0,1}, 1: K={2,3} per step

    // X[m][k] = latent[b][k][jd0+m] ; feature stride = JD floats
    const float* Xb = latent + (size_t)b * (EMB_DIM * JD) + jd0 + m;

    // Preload all 16 A fragments (K = 64 = 16 steps * 4).
    // ISA 32-bit A 16x4 layout: VGPR0 -> K = 2*khalf, VGPR1 -> K = 2*khalf+1.
    v2f a[16];
    #pragma unroll
    for (int s = 0; s < 16; ++s) {
        int k = 4 * s + 2 * khalf;
        a[s].x = Xb[(size_t)k * JD];
        a[s].y = Xb[(size_t)(k + 1) * JD];
    }

    // This lane's output rows' ||x||^2 (C/D layout: VGPR v -> M = v + 8*khalf)
    float xr[8];
    #pragma unroll
    for (int v = 0; v < 8; ++v) xr[v] = xsq[t0 + v + 8 * khalf];

    float bestV[8];
    int   bestI[8];
    #pragma unroll
    for (int v = 0; v < 8; ++v) { bestV[v] = 3.402823466e38f; bestI[v] = 0; }

    #pragma unroll 1
    for (int nt = 0; nt < 8; ++nt) {
        const int N0 = wave * 128 + nt * 16;
        const int n  = N0 + (lane & 15);       // this lane's code column
        const float* Wb = W + (size_t)n * EMB_DIM + 2 * khalf;

        v8f acc = {};
        #pragma unroll
        for (int s = 0; s < 16; ++s) {
            v2f bf;
            bf.x = Wb[4 * s];
            bf.y = Wb[4 * s + 1];
            acc = __builtin_amdgcn_wmma_f32_16x16x4_f32(
                      false, a[s], false, bf, (short)0, acc, false, false);
        }

        const float wn = wsq[n];
        #pragma unroll
        for (int v = 0; v < 8; ++v) {
            int row = t0 + v + 8 * khalf;
            float d = xr[v] + wn - 2.0f * acc[v];
            dist[(size_t)row * NUM_EMB + n] = d;
            // n strictly increases over nt -> strict '<' keeps first occurrence
            if (d < bestV[v]) { bestV[v] = d; bestI[v] = n; }
        }
    }

    // Cross-lane argmin within each 16-lane half (row lives in one vgpr
    // across 16 lanes). Masks < 16 keep the shuffle inside the half.
    #pragma unroll
    for (int off = 8; off >= 1; off >>= 1) {
        #pragma unroll
        for (int v = 0; v < 8; ++v) {
            float ov = __shfl_xor(bestV[v], off, 32);
            int   oi = __shfl_xor(bestI[v], off, 32);
            if (ov < bestV[v] || (ov == bestV[v] && oi < bestI[v])) {
                bestV[v] = ov; bestI[v] = oi;
            }
        }
    }
    if ((lane & 15) == 0) {
        #pragma unroll
        for (int v = 0; v < 8; ++v) {
            int row = v + 8 * khalf;
            lmin[row][wave] = bestV[v];
            lidx[row][wave] = bestI[v];
        }
    }
    __syncthreads();

    // Final per-row reduce over the 8 waves' column slices (wave order is
    // increasing column -> tie-break on smaller index is first-occurrence).
    if (threadIdx.x < 16) {
        int row = threadIdx.x;
        float bv = lmin[row][0];
        int   bi = lidx[row][0];
        #pragma unroll
        for (int w = 1; w < 8; ++w) {
            float ov = lmin[row][w];
            int   oi = lidx[row][w];
            if (ov < bv || (ov == bv && oi < bi)) { bv = ov; bi = oi; }
        }
        idx[t0 + row] = bi;
        atomicAdd(&counts[bi], 1);
    }
}

// ---------------------------------------------------------------------------
// 3) Gather W[idx] into [b,n,j,d] layout (straight-through value == quantized)
//    + commitment-loss partial sums.
// ---------------------------------------------------------------------------
__global__ void __launch_bounds__(256)
vq_gather(const float* __restrict__ latent, const float* __restrict__ W,
          const int* __restrict__ idx, float* __restrict__ qout,
          float* __restrict__ lossacc)
{
    int o  = blockIdx.x * 256 + threadIdx.x;   // index into [b, n, jd]
    int jd = o & 1023;
    int n  = (o >> 10) & 63;
    int b  = o >> 16;
    int t  = b * 1024 + jd;

    float q = W[(size_t)idx[t] * EMB_DIM + n];
    float l = latent[o];
    qout[o] = q;

    float d = q - l;
    float s = d * d;
    #pragma unroll
    for (int off = 16; off >= 1; off >>= 1) s += __shfl_down(s, off, 32);
    if ((threadIdx.x & 31) == 0) atomicAdd(lossacc, s);
}

// ---------------------------------------------------------------------------
// 4) One-hot encodings, float4-vectorized (also performs the zero fill).
// ---------------------------------------------------------------------------
__global__ void __launch_bounds__(256)
vq_enc(const int* __restrict__ idx, float4* __restrict__ enc4)
{
    size_t o4 = (size_t)blockIdx.x * 256 + threadIdx.x;  // float4 index
    int t  = (int)(o4 >> 8);          // 256 float4 per row of 1024
    int k0 = (int)(o4 & 255) * 4;
    int best = idx[t];
    float4 v = make_float4(0.f, 0.f, 0.f, 0.f);
    if (best >= k0 && best < k0 + 4) (&v.x)[best - k0] = 1.0f;
    enc4[o4] = v;
}

// ---------------------------------------------------------------------------
// 5) Loss scaling + perplexity from the code histogram.
// ---------------------------------------------------------------------------
__global__ void __launch_bounds__(1024)
vq_finalize(const int* __restrict__ counts, const float* __restrict__ lossacc,
            float* __restrict__ out_loss, float* __restrict__ out_perp)
{
    __shared__ float red[32];
    int k = threadIdx.x;                       // 1024 threads
    float p    = (float)counts[k] * (1.0f / (float)T_TOTAL);
    float term = p * logf(p + 1e-10f);
    #pragma unroll
    for (int off = 16; off >= 1; off >>= 1) term += __shfl_down(term, off, 32);
    int lane = k & 31, wv = k >> 5;
    if (lane == 0) red[wv] = term;
    __syncthreads();
    if (k < 32) {
        float s = red[k];
        #pragma unroll
        for (int off = 16; off >= 1; off >>= 1) s += __shfl_down(s, off, 32);
        if (k == 0) {
            *out_perp = expf(-s);
            *out_loss = 0.25f * (*lossacc) * (1.0f / (float)Q_ELEMS);
        }
    }
}

// ---------------------------------------------------------------------------
extern "C" void kernel_launch(void* const* d_in, const int* in_sizes, int n_in,
                              void* d_out, int out_size, void* d_ws, size_t ws_size,
                              hipStream_t stream)
{
    (void)in_sizes; (void)n_in; (void)out_size; (void)ws_size;

    const float* latent = (const float*)d_in[0];   // [32, 64, 32, 32]
    const float* W      = (const float*)d_in[1];   // [1024, 64]

    float* out  = (float*)d_out;
    float* qout = out;
    float* loss = out + LOSS_OFF;
    float* perp = out + PERP_OFF;
    float* enc  = out + ENC_OFF;
    float* dist = out + DIST_OFF;

    float* xsq     = (float*)d_ws;                 // 32768 f32
    float* wsq     = xsq + T_TOTAL;                // 1024 f32
    int*   idx     = (int*)(wsq + NUM_EMB);        // 32768 i32
    int*   counts  = idx + T_TOTAL;                // 1024 i32
    float* lossacc = (float*)(counts + NUM_EMB);   // 1 f32

    vq_prep    <<<T_TOTAL / 256,         256, 0, stream>>>(latent, W, xsq, wsq, counts, lossacc);
    vq_dist    <<<T_TOTAL / 16,          256, 0, stream>>>(latent, W, xsq, wsq, dist, idx, counts);
    vq_gather  <<<Q_ELEMS / 256,         256, 0, stream>>>(latent, W, idx, qout, lossacc);
    vq_enc     <<<(ENC_ELEMS / 4) / 256, 256, 0, stream>>>(idx, (float4*)enc);
    vq_finalize<<<1,                    1024, 0, stream>>>(counts, lossacc, loss, perp);
}